// variational_kernel_TT_81226421502509
// MI455X (gfx1250) — compile-verified
//
#include <hip/hip_runtime.h>
#include <hip/hip_bf16.h>

// Problem constants (from the reference):
//   TT_core: (1,200,200,50,32) fp32, K1: 200x200, K2: 200x200, K3: 50x50,
//   indices: (100000,3) ints in [0,50)
#define NB    100000
#define N1    200
#define N2    200
#define N3    50
#define R2    32
#define JKS   (N2 * N3 * R2)   // 320000  (inner size after fixing a)
#define KS    (N3 * R2)        // 1600    (inner size after fixing a,b)
#define TT_ELEMS (N1 * (long long)JKS) // 64,000,000

typedef __attribute__((ext_vector_type(2))) float v2f;
typedef __attribute__((ext_vector_type(4))) float v4f;
typedef __attribute__((ext_vector_type(8))) float v8f;

__device__ __forceinline__ v8f wmma_f32_16x16x4(v2f a, v2f b, v8f c) {
  // V_WMMA_F32_16X16X4_F32: D = A(16x4) x B(4x16) + C(16x16), wave32
  return __builtin_amdgcn_wmma_f32_16x16x4_f32(false, a, false, b, (short)0, c,
                                               false, false);
}

// ---------------------------------------------------------------------------
// Stages 1 & 2: C[b][m,n] = A[m,k] * B[b][k,n]  (row-major everywhere)
//   M = 200, K = 200 (multiple of 8), N = 320000 (stage 1) / 1600 (stage 2).
// One 256-thread block (8 waves) owns one 16-wide N slab and ALL 13 M tiles,
// staging the K x 16 B-slab (12.8 KB) in LDS so each TT/T1 element is read
// from HBM exactly once (b128 loads/stores). A (K1/K2, 160 KB) lives in L2.
// Two independent accumulator chains per wave keep the XDL pipe filled.
// ---------------------------------------------------------------------------
__global__ __launch_bounds__(256) void wmma_gemm_f32(
    const float* __restrict__ A,  // M x K
    const float* __restrict__ B,  // batch stride strideB, each K x N
    float* __restrict__ C,        // batch stride strideC, each M x N
    int M, int K, int N, long long strideB, long long strideC) {
  const long long n0 = (long long)blockIdx.x * 16;
  const float* Bb = B + (long long)blockIdx.z * strideB;
  float* Cb = C + (long long)blockIdx.z * strideC;

  __shared__ float Bs[200 * 16];  // K <= 200

  // Cooperative stage of B[0:K, n0:n0+16] into LDS, float4 granularity:
  // global_load_b128 + ds_store_b128, fully coalesced (4 lanes per 64B row).
  const int total4 = K * 4;  // float4 count
  for (int t = threadIdx.x; t < total4; t += 256) {
    const int kk = t >> 2;
    const int nn = (t & 3) * 4;
    *(v4f*)&Bs[t * 4] = *(const v4f*)&Bb[(long long)kk * N + n0 + nn];
    // Speculative prefetch of the next chunk -> global_prefetch_b8.
    __builtin_prefetch(&Bb[(long long)((t + 256) >> 2) * N + n0], 0, 0);
  }
  __syncthreads();

  const int wave = threadIdx.x >> 5;
  const int lane = threadIdx.x & 31;
  const int lr = lane & 15;  // row/col within tile
  const int lh = lane >> 4;  // half-wave select
  const int mTiles = (M + 15) >> 4;  // 13

  for (int mb = wave; mb < mTiles; mb += 8) {
    const int m0 = mb * 16;
    int arow = m0 + lr;
    if (arow >= M) arow = M - 1;  // clamp: garbage rows computed, never stored
    const float* Arow = A + (long long)arow * K;

    v8f acc0 = {}, acc1 = {};  // two independent WMMA chains
    for (int k = 0; k < K; k += 8) {
      const int kk0 = k + 2 * lh;
      const int kk1 = k + 4 + 2 * lh;
      v2f a0 = *(const v2f*)(Arow + kk0);  // 8B-aligned pair
      v2f a1 = *(const v2f*)(Arow + kk1);
      v2f b0, b1;
      b0.x = Bs[kk0 * 16 + lr];
      b0.y = Bs[(kk0 + 1) * 16 + lr];
      b1.x = Bs[kk1 * 16 + lr];
      b1.y = Bs[(kk1 + 1) * 16 + lr];
      acc0 = wmma_f32_16x16x4(a0, b0, acc0);
      acc1 = wmma_f32_16x16x4(a1, b1, acc1);
    }
    const v8f acc = acc0 + acc1;
#pragma unroll
    for (int v = 0; v < 8; ++v) {
      const int m = m0 + v + 8 * lh;
      if (m < M) Cb[(long long)m * N + n0 + lr] = acc[v];
    }
  }
}

// ---------------------------------------------------------------------------
// Stage 3 (fused): per (a,b) pair: T3 = K3(50x50) @ T2[a,b](50x32),
// then reg = T3 * TT elementwise (identical row-major linear index).
// One block per (a,b); 8 waves = 4 M-tiles x 2 N-tiles sharing the 6.4 KB
// B tile in LDS. Two chains over k=0..47, clamped tail step at k=48.
// ---------------------------------------------------------------------------
__global__ __launch_bounds__(256) void wmma_stage3_fused(
    const float* __restrict__ K3,  // 50 x 50
    const float* __restrict__ T2,  // [ab][50][32]
    const float* __restrict__ TT,  // same layout as T3
    float* __restrict__ T3,        // [ab][50][32]
    float* __restrict__ reg) {
  const int M = 50, K = 50, N = 32;
  const long long ab = blockIdx.x;  // 40000 (a*200 + b)
  const float* Bb = T2 + ab * (long long)KS;

  __shared__ float Bs[N3 * R2];  // 50*32 floats = 6.4 KB
  for (int t = threadIdx.x; t < KS / 4; t += 256)
    *(v4f*)&Bs[t * 4] = *(const v4f*)&Bb[t * 4];
  __syncthreads();

  const int wave = threadIdx.x >> 5;
  const int mb = wave >> 1;  // 0..3 (covers rows 0..63, clamped)
  const int nb = wave & 1;   // 0..1
  const int lane = threadIdx.x & 31;
  const int lr = lane & 15, lh = lane >> 4;

  const int m0 = mb * 16, n0 = nb * 16;
  int arow = m0 + lr;
  if (arow >= M) arow = M - 1;
  const float* Arow = K3 + (long long)arow * K;

  v8f acc0 = {}, acc1 = {};
  for (int k = 0; k < 48; k += 8) {  // kk+1 <= k+7 <= 47 < K: no clamps
    const int kk0 = k + 2 * lh;
    const int kk1 = k + 4 + 2 * lh;
    v2f a0 = *(const v2f*)(Arow + kk0);
    v2f a1 = *(const v2f*)(Arow + kk1);
    v2f b0, b1;
    b0.x = Bs[kk0 * N + n0 + lr];
    b0.y = Bs[(kk0 + 1) * N + n0 + lr];
    b1.x = Bs[kk1 * N + n0 + lr];
    b1.y = Bs[(kk1 + 1) * N + n0 + lr];
    acc0 = wmma_f32_16x16x4(a0, b0, acc0);
    acc1 = wmma_f32_16x16x4(a1, b1, acc1);
  }
  {  // tail k = 48..49 (half-wave with kk=50,51 contributes zeros)
    const int kk = 48 + 2 * lh;
    v2f a, b;
    a.x = (kk < K) ? Arow[kk] : 0.0f;
    a.y = (kk + 1 < K) ? Arow[kk + 1] : 0.0f;
    b.x = (kk < K) ? Bs[kk * N + n0 + lr] : 0.0f;
    b.y = (kk + 1 < K) ? Bs[(kk + 1) * N + n0 + lr] : 0.0f;
    acc0 = wmma_f32_16x16x4(a, b, acc0);
  }
  const v8f acc = acc0 + acc1;

  const long long base = ab * (long long)KS;
#pragma unroll
  for (int v = 0; v < 8; ++v) {
    const int m = m0 + v + 8 * lh;
    if (m < M) {
      const long long idx = base + (long long)m * N + n0 + lr;
      const float t = acc[v];
      T3[idx] = t;
      reg[idx] = t * TT[idx];  // fused elementwise epilogue
    }
  }
}

// ---------------------------------------------------------------------------
// Gather: out[row, 0:32] = T3[i0, i1, i2, 0:32]. One wave per output row,
// lane s loads one float (each row is one contiguous 128B line).
// ---------------------------------------------------------------------------
__global__ __launch_bounds__(256) void gather_rows(
    const float* __restrict__ T3, const int* __restrict__ idx,
    float* __restrict__ out, int nRows) {
  const int row = blockIdx.x * 8 + (threadIdx.x >> 5);
  if (row >= nRows) return;
  const int lane = threadIdx.x & 31;
  const int i0 = idx[row * 3 + 0];
  const int i1 = idx[row * 3 + 1];
  const int i2 = idx[row * 3 + 2];
  const long long base = ((long long)i0 * N2 + i1) * KS + (long long)i2 * R2;
  out[(long long)row * R2 + lane] = T3[base + lane];
}

extern "C" void kernel_launch(void* const* d_in, const int* in_sizes, int n_in,
                              void* d_out, int out_size, void* d_ws,
                              size_t ws_size, hipStream_t stream) {
  const float* TT = (const float*)d_in[0];   // 64M fp32
  const float* K1 = (const float*)d_in[1];   // 200x200
  const float* K2 = (const float*)d_in[2];   // 200x200
  const float* K3 = (const float*)d_in[3];   // 50x50
  const int* indices = (const int*)d_in[4];  // (100000,3)

  float* out = (float*)d_out;                       // 100000*32 floats
  float* reg = (float*)d_out + (long long)NB * R2;  // 64M floats

  // Workspace ping-pong: T1 and T3 share region 0 (T1 dead after stage 2).
  float* T1 = (float*)d_ws;             // 256 MB
  float* T2 = (float*)d_ws + TT_ELEMS;  // 256 MB
  float* T3 = T1;

  // Stage 1: T1 = K1 @ TT   (M=200, K=200, N=320000); 20000 N-slabs.
  wmma_gemm_f32<<<dim3(JKS / 16, 1, 1), 256, 0, stream>>>(
      K1, TT, T1, N1, N1, JKS, 0LL, 0LL);

  // Stage 2: per a: T2[a] = K2 @ T1[a]  (M=200, K=200, N=1600); batch=200.
  wmma_gemm_f32<<<dim3(KS / 16, 1, N1), 256, 0, stream>>>(
      K2, T1, T2, N2, N2, KS, (long long)JKS, (long long)JKS);

  // Stage 3: per (a,b): T3 = K3 @ T2[a,b]; fused reg = T3*TT; batch=40000.
  wmma_stage3_fused<<<dim3(N1 * N2, 1, 1), 256, 0, stream>>>(K3, T2, TT, T3,
                                                             reg);

  // Gather the 100000 requested (a,b,c) rows of T3.
  gather_rows<<<dim3((NB + 7) / 8, 1, 1), 256, 0, stream>>>(T3, indices, out,
                                                            NB);
}